// DISK_22531398435217
// MI455X (gfx1250) — compile-verified
//
#include <hip/hip_runtime.h>
#include <hip/hip_bf16.h>

// ---------------------------------------------------------------------------
// DISK keypoint pipeline for MI455X (gfx1250, wave32)
//   K1: 5x5 NMS (separable max in LDS) + monotonic uint key encode
//   K2: per-batch 4-pass radix select + LDS compaction + bitonic sort
//   K3: descriptor gather -> LDS tile -> row norms via V_WMMA_F32_16X16X4_F32
//       (Gram-matrix diagonal), normalize, write out
// ---------------------------------------------------------------------------

#define DISK_W   768
#define DISK_H   512
#define DISK_HW  (DISK_W * DISK_H)
#define DISK_B   4
#define DISK_C   129
#define DISK_D   128
#define DISK_K   2048
#define CAND     4096

typedef __attribute__((ext_vector_type(2))) float v2f;
typedef __attribute__((ext_vector_type(8))) float v8f;

__device__ __forceinline__ unsigned key_of(float f) {
    unsigned u = __float_as_uint(f);
    return (u & 0x80000000u) ? ~u : (u | 0x80000000u);
}
__device__ __forceinline__ float float_of(unsigned k) {
    unsigned u = (k & 0x80000000u) ? (k & 0x7fffffffu) : ~k;
    return __uint_as_float(u);
}

// ---------------------------------------------------------------------------
// Kernel 1: NMS + key encode.  Tile 32x8 outputs, halo 2, separable max.
// ---------------------------------------------------------------------------
#define TSX 32
#define TSY 8

__global__ __launch_bounds__(256) void disk_nms_encode(
    const float* __restrict__ unet, unsigned* __restrict__ keys)
{
    const int b  = blockIdx.z;
    const int x0 = blockIdx.x * TSX;
    const int y0 = blockIdx.y * TSY;
    const float* __restrict__ hm =
        unet + ((size_t)b * DISK_C + DISK_D) * (size_t)DISK_HW;

    __shared__ float raw[TSY + 4][TSX + 4];   // 12 x 36
    __shared__ float hx [TSY + 4][TSX];       // horizontal 5-max

    const int tid = threadIdx.y * TSX + threadIdx.x;

    for (int i = tid; i < (TSY + 4) * (TSX + 4); i += 256) {
        int r = i / (TSX + 4), c = i % (TSX + 4);
        int gy = y0 + r - 2, gx = x0 + c - 2;
        float v = -3.4028235e38f;
        if (gy >= 0 && gy < DISK_H && gx >= 0 && gx < DISK_W)
            v = hm[(size_t)gy * DISK_W + gx];
        raw[r][c] = v;
    }
    __syncthreads();

    for (int i = tid; i < (TSY + 4) * TSX; i += 256) {
        int r = i / TSX, c = i % TSX;
        float m = raw[r][c];
        m = fmaxf(m, raw[r][c + 1]);
        m = fmaxf(m, raw[r][c + 2]);
        m = fmaxf(m, raw[r][c + 3]);
        m = fmaxf(m, raw[r][c + 4]);
        hx[r][c] = m;
    }
    __syncthreads();

    const int c = threadIdx.x, r = threadIdx.y;
    float m = hx[r][c];
    m = fmaxf(m, hx[r + 1][c]);
    m = fmaxf(m, hx[r + 2][c]);
    m = fmaxf(m, hx[r + 3][c]);
    m = fmaxf(m, hx[r + 4][c]);

    const float h   = raw[r + 2][c + 2];
    const float sup = (h == m) ? h : 0.0f;     // suppress non-maxima
    keys[(size_t)b * DISK_HW + (size_t)(y0 + r) * DISK_W + (x0 + c)] =
        key_of(sup);
}

// ---------------------------------------------------------------------------
// Kernel 2: per-batch exact top-2048.
//   4-pass MSB radix select (8-bit digits, LDS histogram) -> 32-bit threshold
//   compact keys >= T into LDS, bitonic sort (key desc, idx asc),
//   emit keypoints + scores to d_out and sorted indices to workspace.
// ---------------------------------------------------------------------------
__global__ __launch_bounds__(1024) void disk_select_sort(
    const unsigned* __restrict__ keys,
    float* __restrict__ out_kp,      // [B][K][2]
    float* __restrict__ out_sc,      // [B][K]
    int*   __restrict__ ws_idx)      // [B][K]
{
    const int b   = blockIdx.x;
    const int tid = threadIdx.x;
    const unsigned* __restrict__ kb = keys + (size_t)b * DISK_HW;

    __shared__ unsigned hist[256];
    __shared__ unsigned s_prefix;
    __shared__ unsigned s_krem;
    __shared__ unsigned s_cnt;
    __shared__ unsigned ckey[CAND];
    __shared__ int      cidx[CAND];

    if (tid == 0) { s_prefix = 0u; s_krem = DISK_K; }

    // ---- radix select: find exact 2048th-largest 32-bit key ----
    for (int pass = 0; pass < 4; ++pass) {
        for (int i = tid; i < 256; i += 1024) hist[i] = 0u;
        __syncthreads();
        const unsigned prefix = s_prefix;
        const int shift = 24 - 8 * pass;
        for (int i = tid; i < DISK_HW; i += 1024) {
            unsigned key = kb[i];
            bool match = (pass == 0) || ((key >> (shift + 8)) == prefix);
            if (match) atomicAdd(&hist[(key >> shift) & 255u], 1u);
        }
        __syncthreads();
        if (tid == 0) {
            unsigned krem = s_krem, acc = 0u;
            int d = 255;
            for (; d >= 0; --d) {
                unsigned hv = hist[d];
                if (acc + hv >= krem) { s_krem = krem - acc; break; }
                acc += hv;
            }
            if (d < 0) d = 0;  // degenerate safeguard
            s_prefix = (prefix << 8) | (unsigned)d;
        }
        __syncthreads();
    }
    const unsigned T = s_prefix;

    // ---- compaction: all keys >= T (count(>T) < 2048 <= count(>=T)) ----
    for (int i = tid; i < CAND; i += 1024) { ckey[i] = 0u; cidx[i] = 0x7FFFFFFF; }
    if (tid == 0) s_cnt = 0u;
    __syncthreads();
    for (int i = tid; i < DISK_HW; i += 1024) {
        unsigned key = kb[i];
        if (key >= T) {
            unsigned p = atomicAdd(&s_cnt, 1u);
            if (p < CAND) { ckey[p] = key; cidx[p] = i; }
        }
    }
    __syncthreads();

    // ---- bitonic sort: descending key, ascending index (== lax.top_k) ----
    for (unsigned ksz = 2; ksz <= CAND; ksz <<= 1) {
        for (unsigned j = ksz >> 1; j > 0; j >>= 1) {
            for (int t = tid; t < CAND; t += 1024) {
                int p = t ^ (int)j;
                if (p > t) {
                    unsigned ka = ckey[t]; int ia = cidx[t];
                    unsigned kp = ckey[p]; int ip = cidx[p];
                    bool up  = ((t & (int)ksz) == 0);
                    // "x strictly precedes y" in desired final order:
                    bool p_first = (kp > ka) || (kp == ka && ip < ia);
                    bool t_first = (ka > kp) || (ka == kp && ia < ip);
                    if (up ? p_first : t_first) {
                        ckey[t] = kp; cidx[t] = ip;
                        ckey[p] = ka; cidx[p] = ia;
                    }
                }
            }
            __syncthreads();
        }
    }

    // ---- emit keypoints, scores, index list ----
    for (int i = tid; i < DISK_K; i += 1024) {
        unsigned key = ckey[i];
        int idx = cidx[i];
        int x = idx % DISK_W, y = idx / DISK_W;
        size_t o = (size_t)b * DISK_K + i;
        out_kp[o * 2 + 0] = (float)x;
        out_kp[o * 2 + 1] = (float)y;
        out_sc[o]         = float_of(key);
        ws_idx[o]         = (idx >= 0 && idx < DISK_HW) ? idx : 0;
    }
}

// ---------------------------------------------------------------------------
// Kernel 3: gather descriptors + L2 normalize.
//   One wave32 per 16 keypoints.  Squared row norms via the diagonal of the
//   Gram matrix computed with V_WMMA_F32_16X16X4_F32 accumulating over K=128.
//   A-fragment layout (ISA 7.12.2, 32-bit A 16x4): lane<16 holds (M=lane,
//   K=0/1) in v[0]/v[1]; lane>=16 holds (M=lane-16, K=2/3).  With the 32-bit
//   B layout this same register pair serves as B = A^T, so one fragment feeds
//   both operands.  C/D diag (M=N): lane m, vgpr m (m<8); lane m+16, vgpr m-8.
// ---------------------------------------------------------------------------
__global__ __launch_bounds__(128) void disk_gather_norm(
    const float* __restrict__ unet,
    const int*   __restrict__ ws_idx,
    float*       __restrict__ out_desc)   // [B][K][128]
{
    __shared__ float ldsD[4][16][130];    // +2 pad: conflict-free column reads
    __shared__ float sqv [4][32][8];
    __shared__ float inv [4][16];
    __shared__ int   lidx[4][16];

    const int tid  = threadIdx.x;
    const int wave = tid >> 5;
    const int lane = tid & 31;

    const int grp = blockIdx.x * 4 + wave;   // 16 keypoints per group
    const int kg0 = grp * 16;
    const int b   = kg0 >> 11;               // 2048 kp per batch
    const int off = kg0 & 2047;

    if (lane < 16) lidx[wave][lane] = ws_idx[(size_t)b * DISK_K + off + lane];
    __syncthreads();

    const float* __restrict__ base = unet + (size_t)b * DISK_C * (size_t)DISK_HW;

    // phase 1: gather 16x128 tile (one element per lane per iter, coalesced k)
    for (int t = 0; t < 64; ++t) {
        int e = t * 32 + lane;
        int m = e >> 7, k = e & 127;
        int idx = lidx[wave][m];
        ldsD[wave][m][k] = base[(size_t)k * DISK_HW + idx];
    }
    __syncthreads();

    // phase 2: squared row norms = diag(A * A^T) via 32x WMMA f32 16x16x4
    const int m  = lane & 15;
    const int ko = (lane >> 4) << 1;         // 0 for lanes 0-15, 2 for 16-31
    v8f acc = {0.f, 0.f, 0.f, 0.f, 0.f, 0.f, 0.f, 0.f};
#pragma unroll
    for (int c4 = 0; c4 < 32; ++c4) {
        int kk = c4 * 4 + ko;
        v2f a;
        a.x = ldsD[wave][m][kk];
        a.y = ldsD[wave][m][kk + 1];
        acc = __builtin_amdgcn_wmma_f32_16x16x4_f32(
            /*neg_a=*/false, a, /*neg_b=*/false, a,
            /*c_mod=*/(short)0, acc, /*reuse_a=*/false, /*reuse_b=*/false);
    }
#pragma unroll
    for (int r = 0; r < 8; ++r) sqv[wave][lane][r] = acc[r];
    __syncthreads();

    if (lane < 16) {
        int src = (lane < 8) ? lane : lane + 16;
        float s = sqv[wave][src][lane & 7];
        float n = sqrtf(s);
        inv[wave][lane] = 1.0f / fmaxf(n, 1e-12f);
    }
    __syncthreads();

    // phase 3: scaled, fully coalesced write-out
    for (int t = 0; t < 64; ++t) {
        int e = t * 32 + lane;
        int mm = e >> 7, k = e & 127;
        out_desc[((size_t)b * DISK_K + off + mm) * (size_t)DISK_D + k] =
            ldsD[wave][mm][k] * inv[wave][mm];
    }
}

// ---------------------------------------------------------------------------
extern "C" void kernel_launch(void* const* d_in, const int* in_sizes, int n_in,
                              void* d_out, int out_size, void* d_ws, size_t ws_size,
                              hipStream_t stream)
{
    (void)in_sizes; (void)n_in; (void)out_size; (void)ws_size;

    const float* unet = (const float*)d_in[0];

    float* out      = (float*)d_out;
    float* out_kp   = out;                                  // 4*2048*2
    float* out_sc   = out + (size_t)DISK_B * DISK_K * 2;    // 4*2048
    float* out_desc = out_sc + (size_t)DISK_B * DISK_K;     // 4*2048*128

    unsigned* keys = (unsigned*)d_ws;                                   // 6.29 MB
    int* ws_idx    = (int*)((char*)d_ws + (size_t)DISK_B * DISK_HW * 4); // 32 KB

    disk_nms_encode<<<dim3(DISK_W / TSX, DISK_H / TSY, DISK_B),
                      dim3(TSX, TSY), 0, stream>>>(unet, keys);

    disk_select_sort<<<DISK_B, 1024, 0, stream>>>(keys, out_kp, out_sc, ws_idx);

    // 8192 keypoints / (4 waves * 16 kp) = 128 blocks of 128 threads
    disk_gather_norm<<<(DISK_B * DISK_K) / 64, 128, 0, stream>>>(
        unet, ws_idx, out_desc);
}